// ApproxDVS_56796647522832
// MI455X (gfx1250) — compile-verified
//
#include <hip/hip_runtime.h>
#include <hip/hip_bf16.h>

typedef __attribute__((ext_vector_type(2))) float v2f;
typedef __attribute__((ext_vector_type(8))) float v8f;

#define G3      64000            // 40^3 coarse grid points
#define NLM     1024             // landmarks
#define DIM     128
#define VOL     (DIM*DIM*DIM)    // 2097152
#define BETA    0.01f
#define LOG2E   1.4426950408889634f
#define K2      (200.0f*LOG2E)   // (1/alpha)*log2(e), alpha=0.005

__device__ __forceinline__ float axis40(int t) {
    return fmaf((float)t, 2.0f/39.0f, -1.0f);
}

// ---------------------------------------------------------------------------
// Stage A: coarse RBF displacement field via V_WMMA_F32_16X16X4_F32.
// One wave = 16 grid points; 8 waves/block share one LDS-staged landmark set.
// Hot loop: 2x ds_load_b128 + 1x v_wmma + 8x v_exp + packed FMA accumulate.
// Output layout matches reference transpose(3,2,1,0): T[c][k][j][i].
// ---------------------------------------------------------------------------
__global__ void __launch_bounds__(256)
dvs_coarse_wmma(const float* __restrict__ coords,
                const float* __restrict__ offsets,
                float* __restrict__ coarse) {
    __shared__ float4 lmA[NLM];   // (c0, c1, c2, -|c|^2 * K2)
    __shared__ float4 lmB[NLM];   // (vx, vy, vz, 0)

    const int lane = threadIdx.x & 31;
    const int wave = threadIdx.x >> 5;
    const int half = lane >> 4;          // 0: K=0,1  1: K=2,3 (A layout)
    const int lh   = lane & 15;          // M for A, N for B/C/D
    const int b    = blockIdx.y;
    const int tile = blockIdx.x * (blockDim.x >> 5) + wave;   // 0..3999

    // ---- stage landmarks once per block (each loaded exactly once) ----
    {
        const float* cb = coords  + (size_t)b * NLM * 3;
        const float* ob = offsets + (size_t)b * NLM * 3;
        for (int n = threadIdx.x; n < NLM; n += 256) {
            float o0 = ob[n*3+0], o1 = ob[n*3+1], o2 = ob[n*3+2];
            float c0 = cb[n*3+0] + o0;
            float c1 = cb[n*3+1] + o1;
            float c2 = cb[n*3+2] + o2;
            lmA[n] = make_float4(c0, c1, c2, -(c0*c0 + c1*c1 + c2*c2) * K2);
            lmB[n] = make_float4(-o0, -o1, -o2, 0.0f);
        }
    }
    __syncthreads();

    // ---- A matrix: 16 grid points x (gx,gy,gz,0) ----
    int pm = tile * 16 + lh;
    int ia = pm / 1600, ja = (pm / 40) % 40, ka = pm % 40;
    v2f A;
    if (half == 0) { A[0] = axis40(ia); A[1] = axis40(ja); }
    else           { A[0] = axis40(ka); A[1] = 0.0f;       }

    // per-tile |g|^2 * K2 for this lane-half's M values (M = r + 8*half)
    float sgK[8];
#pragma unroll
    for (int r = 0; r < 8; ++r) {
        int p = tile * 16 + r + 8 * half;
        int i = p / 1600, j = (p / 40) % 40, k = p % 40;
        float gx = axis40(i), gy = axis40(j), gz = axis40(k);
        sgK[r] = (gx*gx + gy*gy + gz*gz) * K2;
    }

    float accx[8], accy[8], accz[8], accd[8];
#pragma unroll
    for (int r = 0; r < 8; ++r) { accx[r]=0.f; accy[r]=0.f; accz[r]=0.f; accd[r]=0.f; }

#pragma unroll 2
    for (int chunk = 0; chunk < NLM / 16; ++chunk) {
        int n = chunk * 16 + lh;
        float4 la = lmA[n];                  // ds_load_b128, conflict-free
        float4 lb = lmB[n];                  // ds_load_b128
        v2f Bm;
        if (half == 0) { Bm[0] = la.x; Bm[1] = la.y; }
        else           { Bm[0] = la.z; Bm[1] = 0.0f; }
        v8f zero = {};
        // dot[m][n] = g_m . c_n   (16x16 f32 tile, one instruction)
        v8f dot = __builtin_amdgcn_wmma_f32_16x16x4_f32(
            false, A, false, Bm, (short)0, zero, false, false);
#pragma unroll
        for (int r = 0; r < 8; ++r) {
            // w = exp(-(|g|^2+|c|^2-2 g.c)/alpha) = exp2(2K2*dot + tn - K2|g|^2)
            float w = __builtin_amdgcn_exp2f(fmaf(dot[r], 2.0f * K2, la.w - sgK[r]));
            accd[r] += w;
            accx[r]  = fmaf(w, lb.x, accx[r]);
            accy[r]  = fmaf(w, lb.y, accy[r]);
            accz[r]  = fmaf(w, lb.z, accz[r]);
        }
    }

    // reduce over N (the 16 lanes of each half) with a butterfly
#pragma unroll
    for (int r = 0; r < 8; ++r) {
#pragma unroll
        for (int m = 1; m < 16; m <<= 1) {   // masks < 16 never cross halves
            accd[r] += __shfl_xor(accd[r], m, 32);
            accx[r] += __shfl_xor(accx[r], m, 32);
            accy[r] += __shfl_xor(accy[r], m, 32);
            accz[r] += __shfl_xor(accz[r], m, 32);
        }
    }

    // normalize and scatter into T[c][k][j][i] (transposed coarse field)
    float* outb = coarse + (size_t)b * 3 * G3;
#pragma unroll
    for (int r = 0; r < 8; ++r) {
        if (lh == r) {
            int p = tile * 16 + r + 8 * half;
            int i = p / 1600, j = (p / 40) % 40, k = p % 40;
            int idx = k * 1600 + j * 40 + i;
            float s = (1.0f + BETA) / (accd[r] + BETA);
            outb[0 * G3 + idx] = accx[r] * s;
            outb[1 * G3 + idx] = accy[r] * s;
            outb[2 * G3 + idx] = accz[r] * s;
        }
    }
}

// ---------------------------------------------------------------------------
// Stage B: fused trilinear upsample (40^3 -> 128^3, align_corners=False) +
// identity grid + 3D grid_sample (zeros padding) of the mask volume.
// ---------------------------------------------------------------------------
__device__ __forceinline__ void rsz(int t, int& i0, int& i1, float& f) {
    float c = fmaxf((t + 0.5f) * 0.3125f - 0.5f, 0.0f);   // 40/128
    i0 = (int)c; if (i0 > 39) i0 = 39;
    i1 = i0 + 1; if (i1 > 39) i1 = 39;
    f = c - (float)i0;
}

__global__ void dvs_warp(const float* __restrict__ vol,
                         const float* __restrict__ coarse,
                         float* __restrict__ out, int total) {
    int t = blockIdx.x * blockDim.x + threadIdx.x;
    if (t >= total) return;
    int b   = t >> 21;
    int rem = t & (VOL - 1);
    int X = rem >> 14, Y = (rem >> 7) & 127, Z = rem & 127;

    int x0,x1,y0,y1,z0,z1; float fx,fy,fz;
    rsz(X, x0, x1, fx); rsz(Y, y0, y1, fy); rsz(Z, z0, z1, fz);

    int p0 = x0*1600, p1 = x1*1600, q0 = y0*40, q1 = y1*40;
    float w000=(1.f-fx)*(1.f-fy)*(1.f-fz), w001=(1.f-fx)*(1.f-fy)*fz;
    float w010=(1.f-fx)*fy*(1.f-fz),       w011=(1.f-fx)*fy*fz;
    float w100=fx*(1.f-fy)*(1.f-fz),       w101=fx*(1.f-fy)*fz;
    float w110=fx*fy*(1.f-fz),             w111=fx*fy*fz;

    const float* Tb = coarse + (size_t)b * 3 * G3;
    float F[3];
#pragma unroll
    for (int c = 0; c < 3; ++c) {
        const float* P = Tb + c * G3;
        F[c] = P[p0+q0+z0]*w000 + P[p0+q0+z1]*w001
             + P[p0+q1+z0]*w010 + P[p0+q1+z1]*w011
             + P[p1+q0+z0]*w100 + P[p1+q0+z1]*w101
             + P[p1+q1+z0]*w110 + P[p1+q1+z1]*w111;
    }

    // identity grid (base permuted (0,3,2,1)) + flow
    float gx = fmaf((float)Z, 2.0f/127.0f, -1.0f) + F[0];  // indexes W
    float gy = fmaf((float)Y, 2.0f/127.0f, -1.0f) + F[1];  // indexes H
    float gz = fmaf((float)X, 2.0f/127.0f, -1.0f) + F[2];  // indexes D

    // align_corners=False unnormalize
    float sx = ((gx + 1.0f) * 128.0f - 1.0f) * 0.5f;
    float sy = ((gy + 1.0f) * 128.0f - 1.0f) * 0.5f;
    float sz = ((gz + 1.0f) * 128.0f - 1.0f) * 0.5f;

    float xf = floorf(sx), yf = floorf(sy), zf = floorf(sz);
    float ax = sx - xf, ay = sy - yf, az = sz - zf;
    int ix0 = (int)xf, iy0 = (int)yf, iz0 = (int)zf;

    const float* V = vol + (size_t)b * VOL;
    float acc = 0.0f;
#pragma unroll
    for (int dz = 0; dz < 2; ++dz) {
        int zi = iz0 + dz; float wz = dz ? az : 1.0f - az;
#pragma unroll
        for (int dy = 0; dy < 2; ++dy) {
            int yi = iy0 + dy; float wy = dy ? ay : 1.0f - ay;
#pragma unroll
            for (int dx = 0; dx < 2; ++dx) {
                int xi = ix0 + dx; float wx = dx ? ax : 1.0f - ax;
                if ((unsigned)zi < 128u && (unsigned)yi < 128u && (unsigned)xi < 128u)
                    acc = fmaf(V[zi*16384 + yi*128 + xi], wz*wy*wx, acc);
            }
        }
    }
    out[t] = acc;
}

extern "C" void kernel_launch(void* const* d_in, const int* in_sizes, int n_in,
                              void* d_out, int out_size, void* d_ws, size_t ws_size,
                              hipStream_t stream) {
    const float* mask    = (const float*)d_in[0];  // (2,1,128,128,128)
    const float* coords  = (const float*)d_in[1];  // (2,1024,3)
    const float* offsets = (const float*)d_in[2];  // (2,1024,3)
    float* out    = (float*)d_out;                 // (2,1,128,128,128)
    float* coarse = (float*)d_ws;                  // 2*3*64000 floats = 1.5 MB

    // 4000 tiles/batch, 8 waves (256 thr) per block -> 500 blocks x 2 batches
    dim3 gA(500, 2);
    dvs_coarse_wmma<<<gA, 256, 0, stream>>>(coords, offsets, coarse);

    int total = 2 * VOL;
    dvs_warp<<<(total + 255) / 256, 256, 0, stream>>>(mask, coarse, out, total);
}